// Interp_37563783971448
// MI455X (gfx1250) — compile-verified
//
#include <hip/hip_runtime.h>
#include <hip/hip_bf16.h>

// ---------------------------------------------------------------------------
// Tri-quadratic Lagrange interpolation MSE loss for MI455X (gfx1250, wave32).
//
// Problem shape: N = 4096*512 points, 81^3 fp32 coefficient grid (~2.1 MB,
// L2-resident on a 192 MB L2 part), scalar fp32 output. Gather-latency bound;
// fp32 throughout. WMMA (V_WMMA_F32_16X16X4_F32 with all-ones A) is used as a
// layout-agnostic wave-level sum primitive in the reduction tree.
// ---------------------------------------------------------------------------

typedef __attribute__((ext_vector_type(2))) float v2f;
typedef __attribute__((ext_vector_type(8))) float v8f;

#define GRID_DIM   81
#define STRIDE0    (GRID_DIM * GRID_DIM)   // 6561
#define STRIDE1    GRID_DIM                // 81
#define MESH_N     40.0f
#define BLOCK_SZ   256
#define GRID_SZ    2048                    // N % (BLOCK_SZ*GRID_SZ) == 0 -> uniform loop

// Wave-level fp32 sum. A = all-ones 16x4 -> D[m,n] = sum_k B[k,n] (independent
// of B's exact VGPR striping). b[0] holds the per-lane value, b[1] = 0, so the
// 16 column sums in D's VGPR0 (replicated in lanes 0-15 and 16-31) partition
// the 32 lane values; 4 xor-shuffles finish the sum. EXEC must be all ones.
__device__ __forceinline__ float wave_sum_wmma(float v) {
    v2f a; a[0] = 1.0f; a[1] = 1.0f;
    v2f b; b[0] = v;    b[1] = 0.0f;
    v8f c = {};
    c = __builtin_amdgcn_wmma_f32_16x16x4_f32(
            /*neg_a=*/false, a, /*neg_b=*/false, b,
            /*c_mod=*/(short)0, c, /*reuse_a=*/false, /*reuse_b=*/false);
    float s = c[0];
    s += __shfl_xor(s, 1, 32);
    s += __shfl_xor(s, 2, 32);
    s += __shfl_xor(s, 4, 32);
    s += __shfl_xor(s, 8, 32);
    return s;   // every lane holds the wave total
}

// Quadratic Lagrange weights on nodes {0, 1/2, 1}:
//   L0(t) = (2t-1)(t-1), L1(t) = -4t(t-1), L2(t) = t(2t-1)
__device__ __forceinline__ void dim_setup(float x, int& base, float w[3]) {
    float s = x * MESH_N;
    float cf = floorf(s);
    cf = fminf(fmaxf(cf, 0.0f), MESH_N - 1.0f);
    float t = s - cf;
    float u = 2.0f * t - 1.0f;
    float tm1 = t - 1.0f;
    w[0] = u * tm1;
    w[1] = -4.0f * t * tm1;
    w[2] = t * u;
    base = (int)cf * 2;
}

__global__ void __launch_bounds__(BLOCK_SZ)
interp_loss_partial(const float* __restrict__ pts,
                    const float* __restrict__ coe,
                    float* __restrict__ partial,
                    int n) {
    const int tid    = blockIdx.x * blockDim.x + threadIdx.x;
    const int stride = gridDim.x * blockDim.x;

    float acc = 0.0f;
    for (int i = tid; i < n; i += stride) {
        const float x0 = pts[3 * i + 0];
        const float x1 = pts[3 * i + 1];
        const float x2 = pts[3 * i + 2];

        int b0, b1, b2;
        float w0[3], w1[3], w2[3];
        dim_setup(x0, b0, w0);
        dim_setup(x1, b1, w1);
        dim_setup(x2, b2, w2);

        const float* __restrict__ p = coe + (b0 * STRIDE0 + b1 * STRIDE1 + b2);

        float r = 0.0f;
        #pragma unroll
        for (int ia = 0; ia < 3; ++ia) {
            #pragma unroll
            for (int ib = 0; ib < 3; ++ib) {
                const float* __restrict__ q = p + ia * STRIDE0 + ib * STRIDE1;
                const float wab = w0[ia] * w1[ib];
                // 3 contiguous floats -> compiler can fuse to global_load_b96
                float row = fmaf(w2[0], q[0], fmaf(w2[1], q[1], w2[2] * q[2]));
                r = fmaf(wab, row, r);
            }
        }

        const float tgt = sinf(x0 * 8.0f) + cosf(sqrtf(x1 * 4.0f)) * sinf(x2 * 4.0f);
        const float e = r - tgt;
        acc = fmaf(e, e, acc);
    }

    // Wave reduce (uniform: n divisible by total threads -> EXEC all-1s here)
    float ws = wave_sum_wmma(acc);

    __shared__ float lds[BLOCK_SZ / 32];
    const int lane = threadIdx.x & 31;
    const int wave = threadIdx.x >> 5;
    if (lane == 0) lds[wave] = ws;
    __syncthreads();
    if (threadIdx.x == 0) {
        float s = 0.0f;
        #pragma unroll
        for (int w = 0; w < BLOCK_SZ / 32; ++w) s += lds[w];
        partial[blockIdx.x] = s;
    }
}

__global__ void __launch_bounds__(BLOCK_SZ)
reduce_final(const float* __restrict__ partial, int nparts,
             float* __restrict__ out, float inv_n) {
    float acc = 0.0f;
    for (int i = threadIdx.x; i < nparts; i += blockDim.x)  // nparts % blockDim == 0
        acc += partial[i];

    float ws = wave_sum_wmma(acc);

    __shared__ float lds[BLOCK_SZ / 32];
    const int lane = threadIdx.x & 31;
    const int wave = threadIdx.x >> 5;
    if (lane == 0) lds[wave] = ws;
    __syncthreads();
    if (threadIdx.x == 0) {
        float s = 0.0f;
        #pragma unroll
        for (int w = 0; w < BLOCK_SZ / 32; ++w) s += lds[w];
        out[0] = s * inv_n;
    }
}

extern "C" void kernel_launch(void* const* d_in, const int* in_sizes, int n_in,
                              void* d_out, int out_size, void* d_ws, size_t ws_size,
                              hipStream_t stream) {
    const float* pts = (const float*)d_in[0];   // [4096, 512, 3] fp32
    const float* coe = (const float*)d_in[1];   // [81, 81, 81]  fp32
    float* out     = (float*)d_out;             // scalar fp32 loss
    float* partial = (float*)d_ws;              // GRID_SZ block partials

    const int n = in_sizes[0] / 3;              // 2,097,152 points

    interp_loss_partial<<<GRID_SZ, BLOCK_SZ, 0, stream>>>(pts, coe, partial, n);
    reduce_final<<<1, BLOCK_SZ, 0, stream>>>(partial, GRID_SZ, out, 1.0f / (float)n);
}